// GNNModel_52544629899637
// MI455X (gfx1250) — compile-verified
//
#include <hip/hip_runtime.h>
#include <hip/hip_bf16.h>

#define NCH 64      // channels
#define NGRAPH 64   // graphs
#define LDS_STRIDE 68   // 64 + 4 pad -> conflict-free float2 reads across all 32 lanes

typedef float v2f __attribute__((ext_vector_type(2)));
typedef float v4f __attribute__((ext_vector_type(4)));
typedef float v8f __attribute__((ext_vector_type(8)));

__device__ __forceinline__ float sigmoidf_(float x) { return 1.0f / (1.0f + __expf(-x)); }

// ---------------- utility: zero fill ----------------
__global__ void zero_kernel(float* __restrict__ p, long long n) {
    long long i = (long long)blockIdx.x * blockDim.x + threadIdx.x;
    if (i < n) p[i] = 0.0f;
}

// ---------------- per-graph node counts ----------------
__global__ void counts_kernel(const int* __restrict__ batch, float* __restrict__ counts, int n) {
    int i = blockIdx.x * blockDim.x + threadIdx.x;
    if (i < n) atomicAdd(&counts[batch[i]], 1.0f);
}

// ---------------- dual GEMM: self = h@Wself^T, y = h@Wneigh^T via V_WMMA_F32_16X16X4_F32 ----------------
// Block = 128 threads = 4 waves; each wave computes a 16(node) x 64(out) strip of BOTH GEMMs.
// Weights staged once per block into LDS (padded stride, bank-conflict free); A strip preloaded
// to registers before the barrier so global A traffic overlaps the LDS fill.
__global__ __launch_bounds__(128)
void gemm_dual_kernel(const float* __restrict__ h,
                      const float* __restrict__ Wself,
                      const float* __restrict__ Wneigh,
                      float* __restrict__ outSelf,
                      float* __restrict__ outY,
                      int n) {
    __shared__ float ldsS[NCH * LDS_STRIDE];
    __shared__ float ldsN[NCH * LDS_STRIDE];

    const int wave = threadIdx.x >> 5;
    const int lane = threadIdx.x & 31;
    const int half = lane >> 4;
    const int l16  = lane & 15;
    const int node_base = blockIdx.x * 64 + wave * 16;

    int arow = node_base + l16;
    if (arow >= n) arow = n - 1;                  // clamp loads; keep EXEC all-ones for WMMA
    const float* hrow = h + (size_t)arow * NCH;

    // Preload this wave's A strip: 16 k-steps x float2 (issued before the barrier).
    v2f afrag[16];
#pragma unroll
    for (int s = 0; s < 16; ++s)
        afrag[s] = *(const v2f*)(hrow + s * 4 + half * 2);

    // Cooperative LDS fill of both 64x64 weight matrices (float4 per thread per iter).
#pragma unroll
    for (int i = (int)threadIdx.x; i < 1024; i += 128) {
        int row = i >> 4;
        int c4  = (i & 15) * 4;
        *(v4f*)(&ldsS[row * LDS_STRIDE + c4]) = *(const v4f*)(Wself  + row * NCH + c4);
        *(v4f*)(&ldsN[row * LDS_STRIDE + c4]) = *(const v4f*)(Wneigh + row * NCH + c4);
    }
    __syncthreads();

    v8f accS[4] = {v8f{0}, v8f{0}, v8f{0}, v8f{0}};
    v8f accN[4] = {v8f{0}, v8f{0}, v8f{0}, v8f{0}};

#pragma unroll
    for (int s = 0; s < 16; ++s) {                // K = 64 in steps of 4
        const int ka = s * 4 + half * 2;
#pragma unroll
        for (int t = 0; t < 4; ++t) {             // 4 output-column tiles of 16
            v2f bs = *(const v2f*)(&ldsS[(t * 16 + l16) * LDS_STRIDE + ka]);
            v2f bn = *(const v2f*)(&ldsN[(t * 16 + l16) * LDS_STRIDE + ka]);
            accS[t] = __builtin_amdgcn_wmma_f32_16x16x4_f32(false, afrag[s], false, bs, (short)0, accS[t], false, false);
            accN[t] = __builtin_amdgcn_wmma_f32_16x16x4_f32(false, afrag[s], false, bn, (short)0, accN[t], false, false);
        }
    }

#pragma unroll
    for (int t = 0; t < 4; ++t) {
#pragma unroll
        for (int r = 0; r < 8; ++r) {
            int row = node_base + half * 8 + r;
            if (row < n) {
                size_t off = (size_t)row * NCH + t * 16 + l16;
                outSelf[off] = accS[t][r];
                outY[off]    = accN[t][r];
            }
        }
    }
}

// ---------------- per-node attention scalars (wave per node) ----------------
// adst[n] = dot(lrelu(h[n]), Watt[0:64]); asrc[n] = dot(lrelu(h[n]), Watt[64:128])
__global__ __launch_bounds__(256)
void att_scalar_kernel(const float* __restrict__ h, const float* __restrict__ Watt,
                       float* __restrict__ adst, float* __restrict__ asrc, int n) {
    int node = (int)((blockIdx.x * (long long)blockDim.x + threadIdx.x) >> 5);
    int lane = threadIdx.x & 31;
    if (node >= n) return;
    v2f hv = *(const v2f*)(h + (size_t)node * NCH + lane * 2);
    float l0 = hv.x > 0.0f ? hv.x : 0.2f * hv.x;
    float l1 = hv.y > 0.0f ? hv.y : 0.2f * hv.y;
    float sd = l0 * Watt[lane * 2]       + l1 * Watt[lane * 2 + 1];
    float ss = l0 * Watt[NCH + lane * 2] + l1 * Watt[NCH + lane * 2 + 1];
#pragma unroll
    for (int off = 16; off > 0; off >>= 1) {
        sd += __shfl_down(sd, off, 32);
        ss += __shfl_down(ss, off, 32);
    }
    if (lane == 0) { adst[node] = sd; asrc[node] = ss; }
}

// ---------------- edge scatter (wave per edge) ----------------
__global__ __launch_bounds__(256)
void edge_scatter_kernel(const int* __restrict__ src, const int* __restrict__ dst,
                         const float* __restrict__ y, const float* __restrict__ adst,
                         const float* __restrict__ asrc, float* __restrict__ agg,
                         float* __restrict__ alpha_sum, int ne) {
    int e = (int)((blockIdx.x * (long long)blockDim.x + threadIdx.x) >> 5);
    int lane = threadIdx.x & 31;
    if (e >= ne) return;
    int s = src[e], d = dst[e];
    float alpha = sigmoidf_(adst[d] + asrc[s]);
    v2f yv = *(const v2f*)(y + (size_t)s * NCH + lane * 2);
    float* ag = agg + (size_t)d * NCH + lane * 2;
    atomicAdd(ag,     alpha * yv.x);
    atomicAdd(ag + 1, alpha * yv.y);
    if (lane == 0) atomicAdd(alpha_sum + d, alpha);
}

// ---------------- combine: out = self + agg/max(asum,eps) + b ; (+GN mean accumulation) ----------------
__global__ __launch_bounds__(256)
void combine_kernel(const float* __restrict__ selfb, const float* __restrict__ agg,
                    const float* __restrict__ alpha_sum, const float* __restrict__ bvec,
                    const int* __restrict__ batch, float* __restrict__ outb,
                    float* __restrict__ gsum, int n, int doGN) {
    int idx = blockIdx.x * blockDim.x + threadIdx.x;
    if (idx >= n * 32) return;
    int node = idx >> 5;
    int c = (idx & 31) * 2;
    float as = fmaxf(alpha_sum[node], 1e-6f);
    size_t off = (size_t)node * NCH + c;
    v2f sv = *(const v2f*)(selfb + off);
    v2f av = *(const v2f*)(agg + off);
    float r0 = sv.x + av.x / as + bvec[c];
    float r1 = sv.y + av.y / as + bvec[c + 1];
    outb[off] = r0; outb[off + 1] = r1;
    if (doGN) {
        int g = batch[node];
        atomicAdd(&gsum[g * NCH + c],     r0);
        atomicAdd(&gsum[g * NCH + c + 1], r1);
    }
}

// ---------------- GraphNorm center (in-place) + variance accumulation ----------------
__global__ __launch_bounds__(256)
void gn_center_kernel(float* __restrict__ h, const int* __restrict__ batch,
                      const float* __restrict__ gsum, const float* __restrict__ counts,
                      const float* __restrict__ mscale, float* __restrict__ vsum, int n) {
    int idx = blockIdx.x * blockDim.x + threadIdx.x;
    if (idx >= n * 32) return;
    int node = idx >> 5;
    int c = (idx & 31) * 2;
    int g = batch[node];
    float cnt = fmaxf(counts[g], 1.0f);
    size_t off = (size_t)node * NCH + c;
    v2f hv = *(const v2f*)(h + off);
    float o0 = hv.x - (gsum[g * NCH + c]     / cnt) * mscale[c];
    float o1 = hv.y - (gsum[g * NCH + c + 1] / cnt) * mscale[c + 1];
    h[off] = o0; h[off + 1] = o1;
    atomicAdd(&vsum[g * NCH + c],     o0 * o0);
    atomicAdd(&vsum[g * NCH + c + 1], o1 * o1);
}

// ---------------- GraphNorm scale/shift + KAN ----------------
__global__ __launch_bounds__(256)
void gn_kan_kernel(const float* __restrict__ h2, const int* __restrict__ batch,
                   const float* __restrict__ vsum, const float* __restrict__ counts,
                   const float* __restrict__ gnw, const float* __restrict__ gnb,
                   const float* __restrict__ kanw, const float* __restrict__ kanb,
                   float* __restrict__ hout, int n) {
    int idx = blockIdx.x * blockDim.x + threadIdx.x;
    if (idx >= n * 32) return;
    int node = idx >> 5;
    int c = (idx & 31) * 2;
    int g = batch[node];
    float cnt = fmaxf(counts[g], 1.0f);
    size_t off = (size_t)node * NCH + c;
    v2f hv = *(const v2f*)(h2 + off);
    float oo[2] = {hv.x, hv.y};
    float r[2];
#pragma unroll
    for (int j = 0; j < 2; ++j) {
        int cc = c + j;
        float stdv = sqrtf(vsum[g * NCH + cc] / cnt + 1e-5f);
        float v = gnw[cc] * oo[j] / stdv + gnb[cc];
        // KAN: softmax over 4 basis weights
        float w0 = kanw[cc * 4], w1 = kanw[cc * 4 + 1], w2 = kanw[cc * 4 + 2], w3 = kanw[cc * 4 + 3];
        float m = fmaxf(fmaxf(w0, w1), fmaxf(w2, w3));
        float e0 = __expf(w0 - m), e1 = __expf(w1 - m), e2 = __expf(w2 - m), e3 = __expf(w3 - m);
        float inv = 1.0f / (e0 + e1 + e2 + e3);
        float silu = v * sigmoidf_(v);
        float th   = tanhf(v);
        float sn   = __sinf(v);
        float gs   = __expf(-0.5f * v * v);
        r[j] = (e0 * silu + e1 * th + e2 * sn + e3 * gs) * inv + kanb[cc];
    }
    hout[off] = r[0]; hout[off + 1] = r[1];
}

extern "C" void kernel_launch(void* const* d_in, const int* in_sizes, int n_in,
                              void* d_out, int out_size, void* d_ws, size_t ws_size,
                              hipStream_t stream) {
    const int N = in_sizes[0] / NCH;   // 100000
    const int E = in_sizes[1] / 2;     // 1600000

    const float* x    = (const float*)d_in[0];
    const int* eidx   = (const int*)d_in[1];
    const int* srcI   = eidx;          // row 0 of (2,E)
    const int* dstI   = eidx + E;      // row 1
    const int* batch  = (const int*)d_in[2];

    const float *Wself[3], *Wneigh[3], *Watt[3], *bvec[3];
    const float *gnw[2], *gnb[2], *gms[2], *kanw[2], *kanb[2];
    int pi = 3;
    for (int l = 0; l < 3; ++l) {
        Wself[l]  = (const float*)d_in[pi++];
        Wneigh[l] = (const float*)d_in[pi++];
        Watt[l]   = (const float*)d_in[pi++];
        bvec[l]   = (const float*)d_in[pi++];
        if (l < 2) {
            gnw[l]  = (const float*)d_in[pi++];
            gnb[l]  = (const float*)d_in[pi++];
            gms[l]  = (const float*)d_in[pi++];
            kanw[l] = (const float*)d_in[pi++];
            kanb[l] = (const float*)d_in[pi++];
        }
    }

    // ---- workspace carve ----
    const size_t nc = (size_t)N * NCH;
    size_t need = (4 * nc + 3 * (size_t)N + 2 * (size_t)NGRAPH * NCH + NGRAPH) * sizeof(float);
    if (ws_size < need) return;   // cannot allocate inside graph capture

    float* ws = (float*)d_ws;
    float* B0 = ws;         ws += nc;
    float* B1 = ws;         ws += nc;
    float* B2 = ws;         ws += nc;
    float* B3 = ws;         ws += nc;
    float* alpha_sum = ws;  ws += N;
    float* adst = ws;       ws += N;
    float* asrc = ws;       ws += N;
    float* gsum = ws;       ws += NGRAPH * NCH;
    float* vsum = ws;       ws += NGRAPH * NCH;
    float* counts = ws;     ws += NGRAPH;

    const long long small_span = 3LL * N + 2LL * NGRAPH * NCH;  // alpha_sum..vsum contiguous

    const int tb = 256;
    const int ew_blocks   = (N * 32 + tb - 1) / tb;     // elementwise, thread per (node, 2ch)
    const int gemm_blocks = (N + 63) / 64;
    const int attw_blocks = (N + 7) / 8;                // wave per node
    const int edge_blocks = (E + 7) / 8;                // wave per edge
    const int zero_big    = (int)((nc + tb - 1) / tb);
    const int zero_small  = (int)((small_span + tb - 1) / tb);

    // per-graph node counts (needed by GN layers only; compute once)
    zero_kernel<<<1, 64, 0, stream>>>(counts, NGRAPH);
    counts_kernel<<<(N + tb - 1) / tb, tb, 0, stream>>>(batch, counts, N);

    const float* hin = x;
    for (int l = 0; l < 3; ++l) {
        float *pself, *py, *pagg, *hout;
        if (l == 0)      { pself = B0; py = B1; pagg = B2; hout = B3; }
        else if (l == 1) { pself = B0; py = B1; pagg = B2; hout = B1; }  // y dead after scatter
        else             { pself = B0; py = B2; pagg = B3; hout = nullptr; }

        zero_kernel<<<zero_big,   tb, 0, stream>>>(pagg, (long long)nc);
        zero_kernel<<<zero_small, tb, 0, stream>>>(alpha_sum, small_span);

        gemm_dual_kernel<<<gemm_blocks, 128, 0, stream>>>(hin, Wself[l], Wneigh[l], pself, py, N);
        att_scalar_kernel<<<attw_blocks, tb, 0, stream>>>(hin, Watt[l], adst, asrc, N);
        edge_scatter_kernel<<<edge_blocks, tb, 0, stream>>>(srcI, dstI, py, adst, asrc, pagg, alpha_sum, E);

        float* outb = (l < 2) ? pself : (float*)d_out;
        combine_kernel<<<ew_blocks, tb, 0, stream>>>(pself, pagg, alpha_sum, bvec[l], batch, outb,
                                                     gsum, N, (l < 2) ? 1 : 0);
        if (l < 2) {
            gn_center_kernel<<<ew_blocks, tb, 0, stream>>>(pself, batch, gsum, counts, gms[l], vsum, N);
            gn_kan_kernel<<<ew_blocks, tb, 0, stream>>>(pself, batch, vsum, counts, gnw[l], gnb[l],
                                                        kanw[l], kanb[l], hout, N);
            hin = hout;
        }
    }
}